// VectorQuantizer_EMA_515396076131
// MI455X (gfx1250) — compile-verified
//
#include <hip/hip_runtime.h>
#include <hip/hip_bf16.h>
#include <math.h>

typedef __attribute__((ext_vector_type(2))) float v2f;
typedef __attribute__((ext_vector_type(8))) float v8f;

#define VQ_K 1024
#define VQ_D 64
#define VQ_N 32768            // 32 * 32 * 32
#define DECAY 0.99f
#define ONE_MINUS_DECAY 0.01f
#define EPSV 1e-5f
#define BETA 0.25f

// ---- d_out layout (floats, concatenated in reference return order) ----
#define OUT_LOSS 0
#define OUT_ZQ   1                          // 32*64*32*32 = 2097152
#define OUT_PERP (OUT_ZQ + 2097152)         // 2097153
#define OUT_IDX  (OUT_PERP + 1)             // 2097154, 32768 entries
#define OUT_NCB  (OUT_IDX + VQ_N)           // new_codebook, 65536
#define OUT_NCS  (OUT_NCB + VQ_K * VQ_D)    // new_cluster_size, 1024
#define OUT_NEA  (OUT_NCS + VQ_K)           // new_embed_avg, 65536

// ---- workspace layout (bytes) ----
#define WS_CNORM   0                        // 1024 f32
#define WS_IDX     4096                     // 32768 i32
#define WS_COUNTS  135168                   // 1024 f32   (contiguous zero region start)
#define WS_SUMEMB  139264                   // 65536 f32
#define WS_SCAL    401408                   // 4 f32: [0]=loss_sum [1]=n_sum [2]=plogp_sum
#define WS_ZERO_FLOATS ((401424 - 135168) / 4)  // 66564

// CDNA5 async global->LDS copy: one instruction per wave, 16 B per lane,
// tracked by ASYNCcnt (no VGPR round trip, no LOADcnt/DScnt on the path).
__device__ __forceinline__ void async_copy16(unsigned lds_byte_addr, const float* gaddr) {
    asm volatile("global_load_async_to_lds_b128 %0, %1, off"
                 :: "v"(lds_byte_addr), "v"(gaddr)
                 : "memory");
}

__global__ __launch_bounds__(256) void vq_zero(float* p, int n) {
    int i = blockIdx.x * 256 + threadIdx.x;
    if (i < n) p[i] = 0.0f;
}

__global__ __launch_bounds__(256) void vq_cnorm(const float* __restrict__ codebook,
                                                float* __restrict__ cnorm) {
    int k = blockIdx.x * 256 + threadIdx.x;   // 4 blocks x 256 = 1024
    const float* row = codebook + (k << 6);
    float s = 0.0f;
#pragma unroll
    for (int d = 0; d < VQ_D; ++d) { float x = row[d]; s += x * x; }
    cnorm[k] = s;
}

// One wave per 16-row tile of flat; 8 waves per block; codebook tiles are
// double-buffered in LDS via async global->LDS DMA.
__global__ __launch_bounds__(256) void vq_argmin_wmma(const float* __restrict__ z_e,
                                                      const float* __restrict__ codebook,
                                                      const float* __restrict__ cnorm,
                                                      int* __restrict__ idx_ws,
                                                      float* __restrict__ idx_out) {
    __shared__ __align__(16) float bsh[2 * 16 * 68];  // 2 x (16 codes x 64 f32, pad 68)
    __shared__ __align__(16) float cnsh[VQ_K];        // all codebook norms

    const int lane = threadIdx.x & 31;
    const int wave = threadIdx.x >> 5;
    const int half = lane >> 4;       // 0: K-offset 0,1 / C rows 0-7 ; 1: K-offset 2,3 / rows 8-15
    const int l16  = lane & 15;
    const int kofs = half << 1;
    const int n0   = (blockIdx.x * 8 + wave) * 16;

    // A fragment: lane holds row M=l16, elements (K=kofs, K=kofs+1) per 4-wide K step.
    const int n  = n0 + l16;
    const int b  = n >> 10;
    const int hw = n & 1023;
    const float* zeb = z_e + ((size_t)b << 16) + hw;   // z_e[b][d][hw], element d at zeb[d<<10]

    // per-thread 16B slice of the 4 KB codebook tile (256 threads x 4 floats)
    const int e0 = threadIdx.x << 2;          // 0..1020
    const int rr = e0 >> 6, cc = e0 & 63;
    const float*   gslice = codebook + ((size_t)rr << 6) + cc;
    const unsigned lds0   = (unsigned)(uintptr_t)bsh + (unsigned)((rr * 68 + cc) << 2);
    const unsigned BUFB   = (unsigned)(16 * 68 * 4);

    // stage codebook norms into LDS; kick off async copy of tile 0
    *(float4*)(cnsh + e0) = *(const float4*)(cnorm + e0);
    async_copy16(lds0, gslice);

    v2f afrag[16];
#pragma unroll
    for (int dt = 0; dt < 16; ++dt) {
        int d = (dt << 2) + kofs;
        afrag[dt].x = zeb[(size_t)d << 10];
        afrag[dt].y = zeb[(size_t)(d + 1) << 10];
    }

    float bestv[8];
    int   besti[8];
#pragma unroll
    for (int v = 0; v < 8; ++v) { bestv[v] = 3.4e38f; besti[v] = 0; }

    for (int kt = 0; kt < 64; ++kt) {
        if (kt < 63) {
            // prefetch next tile into the other buffer while we compute this one
            async_copy16(lds0 + (unsigned)((kt + 1) & 1) * BUFB,
                         gslice + (size_t)(kt + 1) * (16 * VQ_D));
            asm volatile("s_wait_asynccnt 0x1" ::: "memory");  // tile kt landed (in-order)
        } else {
            asm volatile("s_wait_asynccnt 0x0" ::: "memory");
        }
        __syncthreads();   // all waves' slices of tile kt are in LDS

        const float* buf = bsh + (kt & 1) * (16 * 68);

        // preload all B fragments, then run the dense WMMA chain
        v2f bfrag[16];
#pragma unroll
        for (int dt = 0; dt < 16; ++dt)
            bfrag[dt] = *(const v2f*)(buf + l16 * 68 + (dt << 2) + kofs);

        v8f acc = {0.f, 0.f, 0.f, 0.f, 0.f, 0.f, 0.f, 0.f};
#pragma unroll
        for (int dt = 0; dt < 16; ++dt)
            acc = __builtin_amdgcn_wmma_f32_16x16x4_f32(
                false, afrag[dt], false, bfrag[dt], (short)0, acc, false, false);

        float cn   = cnsh[(kt << 4) + l16];
        int   kidx = (kt << 4) + l16;
#pragma unroll
        for (int v = 0; v < 8; ++v) {
            float s = cn - 2.0f * acc[v];        // ||c||^2 - 2 a.c  (||a||^2 const per row)
            if (s < bestv[v]) { bestv[v] = s; besti[v] = kidx; }
        }
        __syncthreads();   // everyone done reading buf before it is overwritten
    }

    // reduce across the 16 lanes (N dimension) within each half; first-min tie-break
#pragma unroll
    for (int m = 1; m <= 8; m <<= 1) {
#pragma unroll
        for (int v = 0; v < 8; ++v) {
            float ov = __shfl_xor(bestv[v], m, 32);
            int   oi = __shfl_xor(besti[v], m, 32);
            if (ov < bestv[v] || (ov == bestv[v] && oi < besti[v])) {
                bestv[v] = ov; besti[v] = oi;
            }
        }
    }
    if (l16 == 0) {
#pragma unroll
        for (int v = 0; v < 8; ++v) {
            int nn = n0 + v + (half << 3);
            idx_ws[nn]  = besti[v];
            idx_out[nn] = (float)besti[v];
        }
    }
}

__global__ __launch_bounds__(256) void vq_gather_stats(const float* __restrict__ z_e,
                                                       const float* __restrict__ codebook,
                                                       const int* __restrict__ idx_ws,
                                                       float* __restrict__ zq_out,
                                                       float* __restrict__ counts,
                                                       float* __restrict__ sum_embed,
                                                       float* __restrict__ scalars) {
    int nn = blockIdx.x * 256 + threadIdx.x;
    int b = nn >> 10, hw = nn & 1023;
    int idx = idx_ws[nn];
    const float* zeb = z_e + ((size_t)b << 16) + hw;
    float*       zqb = zq_out + ((size_t)b << 16) + hw;
    const float* cb  = codebook + ((size_t)idx << 6);
    float loss = 0.0f;
#pragma unroll 4
    for (int d = 0; d < VQ_D; ++d) {
        float ze = zeb[(size_t)d << 10];
        float zq = cb[d];
        float diff = zq - ze;
        zqb[(size_t)d << 10] = ze + diff;    // straight-through: ze + (zq - ze)
        loss += diff * diff;
        atomicAdd(&sum_embed[((size_t)idx << 6) + d], ze);
    }
    atomicAdd(&counts[idx], 1.0f);

    __shared__ float red[256];
    red[threadIdx.x] = loss;
    __syncthreads();
    for (int s = 128; s > 0; s >>= 1) {
        if (threadIdx.x < s) red[threadIdx.x] += red[threadIdx.x + s];
        __syncthreads();
    }
    if (threadIdx.x == 0) atomicAdd(&scalars[0], red[0]);
}

__global__ __launch_bounds__(256) void vq_cluster(const float* __restrict__ cluster_size,
                                                  const float* __restrict__ counts,
                                                  float* __restrict__ ncs_out,
                                                  float* __restrict__ scalars) {
    int k = blockIdx.x * 256 + threadIdx.x;   // 4 x 256 = 1024
    float c   = counts[k];
    float ncs = cluster_size[k] * DECAY + ONE_MINUS_DECAY * c;
    ncs_out[k] = ncs;
    float em    = c * (1.0f / (float)VQ_N);
    float plogp = em * logf(em + 1e-10f);

    __shared__ float r1[256];
    __shared__ float r2[256];
    r1[threadIdx.x] = ncs;
    r2[threadIdx.x] = plogp;
    __syncthreads();
    for (int s = 128; s > 0; s >>= 1) {
        if (threadIdx.x < s) {
            r1[threadIdx.x] += r1[threadIdx.x + s];
            r2[threadIdx.x] += r2[threadIdx.x + s];
        }
        __syncthreads();
    }
    if (threadIdx.x == 0) {
        atomicAdd(&scalars[1], r1[0]);   // n = sum(new_cluster_size)
        atomicAdd(&scalars[2], r2[0]);   // sum e_mean*log(e_mean+1e-10)
    }
}

__global__ __launch_bounds__(256) void vq_finalize(const float* __restrict__ embed_avg,
                                                   const float* __restrict__ sum_embed,
                                                   const float* __restrict__ ncs_out,
                                                   const float* __restrict__ scalars,
                                                   float* __restrict__ nea_out,
                                                   float* __restrict__ ncb_out,
                                                   float* __restrict__ loss_out,
                                                   float* __restrict__ perp_out) {
    int i = blockIdx.x * 256 + threadIdx.x;   // 256 x 256 = 65536 = K*D
    int k = i >> 6;
    float nea = embed_avg[i] * DECAY + ONE_MINUS_DECAY * sum_embed[i];
    nea_out[i] = nea;
    float nsum = scalars[1];
    float smoothed = (ncs_out[k] + EPSV) / (nsum + (float)VQ_K * EPSV) * nsum;
    ncb_out[i] = nea / smoothed;
    if (i == 0) {
        loss_out[0] = BETA * scalars[0] / (float)(VQ_N * VQ_D);
        perp_out[0] = expf(-scalars[2]);
    }
}

extern "C" void kernel_launch(void* const* d_in, const int* in_sizes, int n_in,
                              void* d_out, int out_size, void* d_ws, size_t ws_size,
                              hipStream_t stream) {
    const float* z_e          = (const float*)d_in[0];
    const float* codebook     = (const float*)d_in[1];
    const float* cluster_size = (const float*)d_in[2];
    const float* embed_avg    = (const float*)d_in[3];
    float* out = (float*)d_out;

    char* ws = (char*)d_ws;
    float* cnorm     = (float*)(ws + WS_CNORM);
    int*   idx_ws    = (int*)(ws + WS_IDX);
    float* counts    = (float*)(ws + WS_COUNTS);
    float* sum_embed = (float*)(ws + WS_SUMEMB);
    float* scalars   = (float*)(ws + WS_SCAL);

    // 1) zero accumulators (counts | sum_embed | scalars are contiguous)
    vq_zero<<<(WS_ZERO_FLOATS + 255) / 256, 256, 0, stream>>>(counts, WS_ZERO_FLOATS);

    // 2) codebook squared norms
    vq_cnorm<<<VQ_K / 256, 256, 0, stream>>>(codebook, cnorm);

    // 3) argmin via f32 WMMA + async double-buffered LDS staging
    vq_argmin_wmma<<<(VQ_N / 16) / 8, 256, 0, stream>>>(z_e, codebook, cnorm,
                                                        idx_ws, out + OUT_IDX);

    // 4) gather z_q (straight-through), loss partials, counts, sum_embed
    vq_gather_stats<<<VQ_N / 256, 256, 0, stream>>>(z_e, codebook, idx_ws,
                                                    out + OUT_ZQ, counts, sum_embed, scalars);

    // 5) EMA cluster sizes + scalar reductions
    vq_cluster<<<VQ_K / 256, 256, 0, stream>>>(cluster_size, counts, out + OUT_NCS, scalars);

    // 6) EMA embed_avg, codebook normalization, loss, perplexity
    vq_finalize<<<(VQ_K * VQ_D) / 256, 256, 0, stream>>>(embed_avg, sum_embed,
                                                         out + OUT_NCS, scalars,
                                                         out + OUT_NEA, out + OUT_NCB,
                                                         out + OUT_LOSS, out + OUT_PERP);
}